// LowpassFilter_33311766347840
// MI455X (gfx1250) — compile-verified
//
#include <hip/hip_runtime.h>
#include <math.h>

// ---------------- problem constants ----------------
#define WLEN   1022                       // FFT / window length
#define HOPX   256
#define NFRM   64                         // frames
#define BSZ    64                         // batch
#define NBIN   512                        // rfft bins (W/2+1)
#define TLEN   (HOPX*(NFRM-1)+WLEN)       // 17150
#define PADL   18                         // 3*max(len(a),len(b))
#define TE     (TLEN + 2*PADL)            // 17186
#define NCOL   (BSZ*NFRM)                 // 4096 GEMM columns
#define KDIM   1024                       // padded K (and padded M for gemm1)
#define MTILES 64                         // 1024/16
#define KTILES 256                        // 1024/4
#define KCH    16                         // ktiles staged per LDS chunk
#define NCHUNK (KTILES/KCH)               // 16 chunks
#define LDS_CHUNK_FL (4*KCH*64)           // 4 N-tiles * 16 kt * 64 floats = 4096 floats
#define TWO_PI 6.28318530717958647692f

typedef __attribute__((ext_vector_type(2))) float v2f;
typedef __attribute__((ext_vector_type(4))) float v4f;
typedef __attribute__((ext_vector_type(8))) float v8f;
typedef __attribute__((ext_vector_type(4))) int   v4i;

// Butterworth(5, 0.5) via bilinear transform (computed analytically):
//   b = kd*[1,5,10,10,5,1], kd = 1/(2*(2+2cos36)*(2+2cos72)) = 0.0527864045
//   a = [1, 0, 6-2*sqrt5-2/sqrt5, 0, 9-4*sqrt5, 0]
__device__ __constant__ float c_B[6] = {0.0527864045f, 0.2639320225f, 0.5278640450f,
                                        0.5278640450f, 0.2639320225f, 0.0527864045f};
__device__ __constant__ float c_A[6] = {1.0f, 0.0f, 0.6334368540f, 0.0f, 0.0557280900f, 0.0f};
// lfilter_zi steady-state (hand-solved back-substitution of (I - A^T) zi = Bv)
__device__ __constant__ float c_ZI[5] = {0.9472135955f, 0.6832815730f, 0.7888543820f,
                                         0.2609903369f, 0.0527864045f};

static __device__ inline float hannw(int i) {
    return 0.5f - 0.5f * __cosf(TWO_PI * (float)i / (float)WLEN);
}

// ---------------- fp32 WMMA wrapper (V_WMMA_F32_16X16X4_F32) ----------------
static __device__ inline v8f wmma_f32_16x16x4(v2f a, v2f b, v8f c) {
    return __builtin_amdgcn_wmma_f32_16x16x4_f32(false, a, false, b, (short)0, c, false, false);
}

// ---------------- async global->LDS staging helper ----------------
#if __has_builtin(__builtin_amdgcn_global_load_async_to_lds_b128)
#define HAVE_ASYNC_LDS 1
#else
#define HAVE_ASYNC_LDS 0
#endif

static __device__ inline void stage_chunk(const float* __restrict__ Bfrag, float* __restrict__ buf,
                                          int Ngrp, int c, int tid) {
    // chunk = 4 contiguous 4KB global segments (one per N-tile), fragment order.
    int t = tid >> 6;                      // N-tile 0..3
    int li = tid & 63;
    const float* gbase = Bfrag + ((size_t)(Ngrp * 4 + t) * KTILES + c * KCH) * 64;
    float* lbase = buf + t * (KCH * 64);
    #pragma unroll
    for (int q = 0; q < 4; ++q) {
        int i4 = li + 64 * q;              // float4 index within tile region [0,256)
#if HAVE_ASYNC_LDS
        __builtin_amdgcn_global_load_async_to_lds_b128(
            (__attribute__((address_space(1))) v4i*)(gbase + i4 * 4),
            (__attribute__((address_space(3))) v4i*)(lbase + i4 * 4), 0, 0);
#else
        v4f v = *(const v4f*)(gbase + i4 * 4);
        *(v4f*)(lbase + i4 * 4) = v;
#endif
    }
}

static __device__ inline void wait_stage() {
#if HAVE_ASYNC_LDS
    asm volatile("s_wait_asynccnt 0" ::: "memory");
#endif
}

// ---------------- core: one block = 8 waves (8 M-tiles) x shared 64-col B strip ----------------
// LDS double buffer: 2 * 4096 floats = 32 KB. B read once per block from L2 (8x traffic cut).
static __device__ inline void gemm_core(const float* __restrict__ Afrag,
                                        const float* __restrict__ Bfrag,
                                        float* __restrict__ ldsB,
                                        int Mtile, int Ngrp, int lane, int tid, v8f acc[4]) {
    const v2f* Af = (const v2f*)Afrag;
    int abase = Mtile * KTILES * 32 + lane;

    stage_chunk(Bfrag, ldsB, Ngrp, 0, tid);
    wait_stage();
    __syncthreads();

    for (int c = 0; c < NCHUNK; ++c) {
        if (c + 1 < NCHUNK)
            stage_chunk(Bfrag, ldsB + ((c + 1) & 1) * LDS_CHUNK_FL, Ngrp, c + 1, tid);
        const v2f* Lb = (const v2f*)(ldsB + (c & 1) * LDS_CHUNK_FL);
        #pragma unroll 4
        for (int kl = 0; kl < KCH; ++kl) {
            v2f a  = Af[abase + (c * KCH + kl) * 32];
            v2f b0 = Lb[0 * (KCH * 32) + kl * 32 + lane];
            v2f b1 = Lb[1 * (KCH * 32) + kl * 32 + lane];
            v2f b2 = Lb[2 * (KCH * 32) + kl * 32 + lane];
            v2f b3 = Lb[3 * (KCH * 32) + kl * 32 + lane];
            acc[0] = wmma_f32_16x16x4(a, b0, acc[0]);
            acc[1] = wmma_f32_16x16x4(a, b1, acc[1]);
            acc[2] = wmma_f32_16x16x4(a, b2, acc[2]);
            acc[3] = wmma_f32_16x16x4(a, b3, acc[3]);
        }
        wait_stage();
        __syncthreads();   // all staging landed; all waves done with current buffer
    }
}

// ---------------- utility: zero a float buffer ----------------
__global__ void k_zero(float* __restrict__ p, int n) {
    int i = blockIdx.x * blockDim.x + threadIdx.x;
    if (i < n) p[i] = 0.0f;
}

// ---------------- synthesis window: hann / overlap-den ----------------
__global__ void k_synwin(float* __restrict__ syn) {
    __shared__ float den_s[HOPX];
    int j = threadIdx.x;                       // 0..255
    float d = 0.0f;
    #pragma unroll
    for (int o = 0; o < 4; ++o) {
        int i = j + HOPX * o;
        if (i < WLEN) { float fw = hannw(i); d += fw * fw; }
    }
    den_s[j] = d;
    __syncthreads();
    for (int n = j; n < WLEN; n += HOPX)       // n % 256 == j
        syn[n] = hannw(n) / den_s[j];
}

// ---------------- generate DFT operand A in WMMA fragment order ----------------
// Fragment order: [Mtile(64)][ktile(256)][lane(32)][2], lane: sub=lane&15, hi=lane>>4
//   A[m][k] with m = Mtile*16+sub, k = ktile*4 + 2*hi + j   (ISA 16x4 f32 A layout)
__global__ void k_genA(float* __restrict__ A, int mode) {
    int idx = blockIdx.x * blockDim.x + threadIdx.x;      // 2^20 total
    int j    = idx & 1;
    int lane = (idx >> 1) & 31;
    int kt   = (idx >> 6) & 255;
    int Mt   = idx >> 14;
    int m = Mt * 16 + (lane & 15);
    int k = kt * 4 + 2 * (lane >> 4) + j;
    float val = 0.0f;
    if (mode == 0) {
        int n = m;                                        // time row (valid < 1022)
        if (n < WLEN) {
            int kk = (k < NBIN) ? k : (k - NBIN);
            float w = (kk == 0 || kk == NBIN - 1) ? 1.0f : 2.0f;
            float ang = TWO_PI * (float)((kk * n) % WLEN) / (float)WLEN;
            float tr = (k < NBIN) ? __cosf(ang) : -__sinf(ang);
            val = w * tr * (1.0f / (float)WLEN);
        }
    } else {
        int n = k;                                        // time column (pad >= 1022 -> 0)
        if (n < WLEN) {
            int mm = (m < NBIN) ? m : (m - NBIN);
            float ang = TWO_PI * (float)((mm * n) % WLEN) / (float)WLEN;
            val = (m < NBIN) ? __cosf(ang) : -__sinf(ang);
        }
    }
    A[idx] = val;
}

// ---------------- pack GEMM1 B = [Xr ; Xi] (1024 x 4096) in fragment order ----------------
__global__ void k_packB1(const float* __restrict__ x, float* __restrict__ B) {
    int idx = blockIdx.x * blockDim.x + threadIdx.x;      // 2^22 total
    int j    = idx & 1;
    int lane = (idx >> 1) & 31;
    int kt   = (idx >> 6) & 255;
    int Nt   = idx >> 14;
    int N = Nt * 16 + (lane & 15);
    int b = N >> 6, f = N & 63;
    int k = kt * 4 + 2 * (lane >> 4) + j;
    float val;
    if (k < NBIN) val = x[((b * NBIN + k) * NFRM + f) * 2 + 0];
    else          val = x[((b * NBIN + (k - NBIN)) * NFRM + f) * 2 + 1];
    B[idx] = val;
}

// ---------------- pack GEMM2 B = windowed frames (1024 x 4096, rows>=1022 zero) ----------------
__global__ void k_packB2(const float* __restrict__ filt, float* __restrict__ B) {
    int idx = blockIdx.x * blockDim.x + threadIdx.x;
    int j    = idx & 1;
    int lane = (idx >> 1) & 31;
    int kt   = (idx >> 6) & 255;
    int Nt   = idx >> 14;
    int N = Nt * 16 + (lane & 15);
    int b = N >> 6, f = N & 63;
    int n = kt * 4 + 2 * (lane >> 4) + j;                 // time within frame
    float val = 0.0f;
    if (n < WLEN) val = filt[b * TLEN + f * HOPX + n] * hannw(n);
    B[idx] = val;
}

// ---------------- GEMM1: iDFT + synth window + overlap-add (atomic) ----------------
__global__ void k_gemm1(const float* __restrict__ Afrag, const float* __restrict__ Bfrag,
                        const float* __restrict__ syn, float* __restrict__ signals) {
    __shared__ __align__(16) float ldsB[2 * LDS_CHUNK_FL];
    int tid  = threadIdx.x;
    int lane = tid & 31;
    int w    = tid >> 5;
    int Ngrp  = blockIdx.x & 63;
    int Mtile = (blockIdx.x >> 6) * 8 + w;
    v8f acc[4] = {};
    gemm_core(Afrag, Bfrag, ldsB, Mtile, Ngrp, lane, tid, acc);
    int sub = lane & 15, hi = lane >> 4;
    #pragma unroll
    for (int t = 0; t < 4; ++t) {
        int N = (Ngrp * 4 + t) * 16 + sub;
        int b = N >> 6, f = N & 63;
        float* sp = signals + b * TLEN + f * HOPX;
        #pragma unroll
        for (int r = 0; r < 8; ++r) {
            int n = Mtile * 16 + 8 * hi + r;              // time within frame
            if (n < WLEN) atomicAdd(sp + n, acc[t][r] * syn[n]);
        }
    }
}

// ---------------- GEMM2: fDFT -> output (b, bin, f, {re,im}) ----------------
__global__ void k_gemm2(const float* __restrict__ Afrag, const float* __restrict__ Bfrag,
                        float* __restrict__ out) {
    __shared__ __align__(16) float ldsB[2 * LDS_CHUNK_FL];
    int tid  = threadIdx.x;
    int lane = tid & 31;
    int w    = tid >> 5;
    int Ngrp  = blockIdx.x & 63;
    int Mtile = (blockIdx.x >> 6) * 8 + w;
    v8f acc[4] = {};
    gemm_core(Afrag, Bfrag, ldsB, Mtile, Ngrp, lane, tid, acc);
    int sub = lane & 15, hi = lane >> 4;
    #pragma unroll
    for (int t = 0; t < 4; ++t) {
        int N = (Ngrp * 4 + t) * 16 + sub;
        int b = N >> 6, f = N & 63;
        #pragma unroll
        for (int r = 0; r < 8; ++r) {
            int m = Mtile * 16 + 8 * hi + r;              // 0..511 = Re, 512..1023 = Im
            int mm = (m < NBIN) ? m : (m - NBIN);
            int c  = (m < NBIN) ? 0 : 1;
            out[((b * NBIN + mm) * NFRM + f) * 2 + c] = acc[t][r];
        }
    }
}

// ---------------- filtfilt: one lane per batch, 2 sequential IIR passes ----------------
static __device__ inline float xe_at(const float* __restrict__ x, int i) {
    int t = i - PADL;
    if (t < 0)      return 2.0f * x[0]        - x[-t];
    if (t >= TLEN)  return 2.0f * x[TLEN - 1] - x[2 * TLEN - 2 - t];
    return x[t];
}

__global__ void k_filtfilt(float* __restrict__ signals, float* __restrict__ ybuf) {
    int b = threadIdx.x;                       // 0..63, one batch per lane
    const float* x = signals + b * TLEN;
    float b0 = c_B[0], b1 = c_B[1], b2 = c_B[2], b3 = c_B[3], b4 = c_B[4], b5 = c_B[5];
    float a1 = c_A[1], a2 = c_A[2], a3 = c_A[3], a4 = c_A[4], a5 = c_A[5];

    float x0 = xe_at(x, 0);
    float z0 = c_ZI[0] * x0, z1 = c_ZI[1] * x0, z2 = c_ZI[2] * x0,
          z3 = c_ZI[3] * x0, z4 = c_ZI[4] * x0;
    for (int t = 0; t < TE; ++t) {
        float xt = xe_at(x, t);
        float y  = fmaf(b0, xt, z0);
        z0 = fmaf(b1, xt, z1) - a1 * y;
        z1 = fmaf(b2, xt, z2) - a2 * y;
        z2 = fmaf(b3, xt, z3) - a3 * y;
        z3 = fmaf(b4, xt, z4) - a4 * y;
        z4 = b5 * xt - a5 * y;
        ybuf[t * BSZ + b] = y;                 // lane-coalesced
    }
    float y0 = ybuf[(TE - 1) * BSZ + b];
    z0 = c_ZI[0] * y0; z1 = c_ZI[1] * y0; z2 = c_ZI[2] * y0;
    z3 = c_ZI[3] * y0; z4 = c_ZI[4] * y0;
    for (int t = 0; t < TE; ++t) {
        int idx = TE - 1 - t;
        float xt = ybuf[idx * BSZ + b];
        float y  = fmaf(b0, xt, z0);
        z0 = fmaf(b1, xt, z1) - a1 * y;
        z1 = fmaf(b2, xt, z2) - a2 * y;
        z2 = fmaf(b3, xt, z3) - a3 * y;
        z3 = fmaf(b4, xt, z4) - a4 * y;
        z4 = b5 * xt - a5 * y;
        if (idx >= PADL && idx < TE - PADL)
            signals[b * TLEN + (idx - PADL)] = y;   // second reverse folded into index
    }
}

// ---------------- launch ----------------
extern "C" void kernel_launch(void* const* d_in, const int* in_sizes, int n_in,
                              void* d_out, int out_size, void* d_ws, size_t ws_size,
                              hipStream_t stream) {
    (void)in_sizes; (void)n_in; (void)out_size; (void)ws_size;
    const float* x = (const float*)d_in[0];
    float* out = (float*)d_out;
    float* ws  = (float*)d_ws;

    size_t o = 0;
    float* Afrag   = ws + o; o += (size_t)1024 * 1024;          // 4 MB, regenerated per GEMM
    float* Bfrag   = ws + o; o += (size_t)KDIM * NCOL;          // 16 MB, reused per GEMM
    float* signals = ws + o; o += (size_t)BSZ * TLEN;           // OLA accumulator -> filt output
    float* ybuf    = ws + o; o += (size_t)BSZ * TE;             // IIR intermediate
    float* synwin  = ws + o; o += 1024;                         // total ~28.4 MB

    const int NS = BSZ * TLEN;
    k_zero    <<<(NS + 255) / 256, 256, 0, stream>>>(signals, NS);
    k_synwin  <<<1, HOPX, 0, stream>>>(synwin);
    k_genA    <<<4096, 256, 0, stream>>>(Afrag, 0);
    k_packB1  <<<16384, 256, 0, stream>>>(x, Bfrag);
    k_gemm1   <<<512, 256, 0, stream>>>(Afrag, Bfrag, synwin, signals);
    k_filtfilt<<<1, BSZ, 0, stream>>>(signals, ybuf);
    k_genA    <<<4096, 256, 0, stream>>>(Afrag, 1);
    k_packB2  <<<16384, 256, 0, stream>>>(signals, Bfrag);
    k_gemm2   <<<512, 256, 0, stream>>>(Afrag, Bfrag, out);
}